// LIFAct_6244882448891
// MI455X (gfx1250) — compile-verified
//
#include <hip/hip_runtime.h>
#include <hip/hip_bf16.h>
#include <stdint.h>

// LIF forward scan, T=4, fp32. Memory-bound: 268 MB traffic -> ~11.5us floor
// at 23.3 TB/s. Data path: global_load_async_to_lds_b128 in GVS mode
// (SGPR base + 32-bit lane offset, th:TH_LOAD_NT) -> s_wait_asynccnt ->
// ds_load_b128 readback -> VALU scan -> NT b128 stores with saddr+voffset.

typedef float v4f __attribute__((ext_vector_type(4)));

#define THREADS 256
#define T_STEPS 4
#define TILE_F (THREADS * 4)   // floats per time slice per block (1024)

__global__ __launch_bounds__(THREADS)
void lif_scan_kernel(const float* __restrict__ x, float* __restrict__ out,
                     int n4 /* float4s per time slice */,
                     int nf /* floats per time slice */) {
  __shared__ __align__(16) float tile[T_STEPS * TILE_F];   // 16 KB

  const int tid = threadIdx.x;
  const int g4  = blockIdx.x * THREADS + tid;              // float4 index in slice
  if (g4 >= n4) return;

  // Per-lane LDS byte address of this lane's 16B slot for t=0 (low 32 bits of
  // the generic pointer are the LDS byte offset; ptrtoint escapes `tile` so
  // the "memory"-clobbered asm below is known to touch it).
  const uint32_t lbase = (uint32_t)(uintptr_t)(&tile[0]) + (uint32_t)(tid * 16);

  // GVS addressing: uniform 64-bit slice base in SGPRs, per-lane 32-bit byte
  // offset in one VGPR (max ~33.5 MB, fits signed 32-bit).
  const uint32_t voff    = (uint32_t)g4 * 16u;             // byte offset in slice
  const uint64_t sbase   = (uint64_t)(uintptr_t)x;
  const uint64_t gstride = (uint64_t)nf * 4ull;            // bytes between slices

  // Four independent async 16B copies: HBM -> LDS, non-temporal (read-once
  // stream larger than the 192 MB L2).
  asm volatile("global_load_async_to_lds_b128 %0, %1, %2 th:TH_LOAD_NT"
      :: "v"(lbase),                    "v"(voff), "s"(sbase)                 : "memory");
  asm volatile("global_load_async_to_lds_b128 %0, %1, %2 th:TH_LOAD_NT"
      :: "v"(lbase + 1u * TILE_F * 4u), "v"(voff), "s"(sbase + 1ull * gstride) : "memory");
  asm volatile("global_load_async_to_lds_b128 %0, %1, %2 th:TH_LOAD_NT"
      :: "v"(lbase + 2u * TILE_F * 4u), "v"(voff), "s"(sbase + 2ull * gstride) : "memory");
  asm volatile("global_load_async_to_lds_b128 %0, %1, %2 th:TH_LOAD_NT"
      :: "v"(lbase + 3u * TILE_F * 4u), "v"(voff), "s"(sbase + 3ull * gstride) : "memory");
  asm volatile("s_wait_asynccnt 0" ::: "memory");

  // Each lane reads back only the data it loaded -> no barrier needed.
  const v4f* lp = (const v4f*)(&tile[tid * 4]);
  v4f x0 = lp[0 * THREADS];
  v4f x1 = lp[1 * THREADS];
  v4f x2 = lp[2 * THREADS];
  v4f x3 = lp[3 * THREADS];

  // LIF scan: mem = mem*0.25 + x_t; spike = mem > 0.5; mem = spike ? 0 : mem.
  v4f o0, o1, o2, o3;
#pragma unroll
  for (int c = 0; c < 4; ++c) {
    float mem = x0[c];                       // mem starts at 0
    o0[c] = (mem > 0.5f) ? 1.0f : 0.0f;
    mem   = (mem > 0.5f) ? 0.0f : mem;

    mem = mem * 0.25f + x1[c];
    o1[c] = (mem > 0.5f) ? 1.0f : 0.0f;
    mem   = (mem > 0.5f) ? 0.0f : mem;

    mem = mem * 0.25f + x2[c];
    o2[c] = (mem > 0.5f) ? 1.0f : 0.0f;
    mem   = (mem > 0.5f) ? 0.0f : mem;

    mem = mem * 0.25f + x3[c];
    o3[c] = (mem > 0.5f) ? 1.0f : 0.0f;
  }

  // Streaming output, written once, never re-read: non-temporal b128 stores.
  // 32-bit index math off the uniform base -> saddr + voffset addressing.
  const uint32_t onf  = (uint32_t)nf;
  const uint32_t obase = (uint32_t)g4 * 4u;
  __builtin_nontemporal_store(o0, (v4f*)(out + obase));
  __builtin_nontemporal_store(o1, (v4f*)(out + obase + 1u * onf));
  __builtin_nontemporal_store(o2, (v4f*)(out + obase + 2u * onf));
  __builtin_nontemporal_store(o3, (v4f*)(out + obase + 3u * onf));
}

extern "C" void kernel_launch(void* const* d_in, const int* in_sizes, int n_in,
                              void* d_out, int out_size, void* d_ws, size_t ws_size,
                              hipStream_t stream) {
  (void)n_in; (void)out_size; (void)d_ws; (void)ws_size;

  const float* x = (const float*)d_in[0];
  float* out     = (float*)d_out;

  const int total = in_sizes[0];        // T * B*C*H*W = 33,554,432
  const int nf    = total / T_STEPS;    // floats per time slice = 8,388,608
  const int n4    = nf / 4;             // float4s per slice     = 2,097,152

  const int blocks = (n4 + THREADS - 1) / THREADS;   // 8192
  hipLaunchKernelGGL(lif_scan_kernel, dim3(blocks), dim3(THREADS), 0, stream,
                     x, out, n4, nf);
}